// QuantumDistiller_64922725646997
// MI455X (gfx1250) — compile-verified
//
#include <hip/hip_runtime.h>
#include <hip/hip_bf16.h>

typedef __attribute__((ext_vector_type(2))) float v2f;
typedef __attribute__((ext_vector_type(8))) float v8f;

#define HW 65536          // 256*256
#define NC 64             // channels C == outputs O
#define NB 16             // batch
#define OUT_MAIN (16ll * 64 * 65536)   // 67,108,864

// ---------------------------------------------------------------------------
// Kernel 1: per-(b,c) spatial mean.  grid = B*C blocks of 256 threads.
// ---------------------------------------------------------------------------
__global__ __launch_bounds__(256) void qd_mean_kernel(const float* __restrict__ x,
                                                      float* __restrict__ xg) {
  const int j = blockIdx.x;                       // j = b*64 + c
  const float4* p = (const float4*)(x + (long)j * HW);
  float s = 0.f;
  for (int i = threadIdx.x; i < HW / 4; i += 256) {
    float4 v = p[i];
    s += (v.x + v.y) + (v.z + v.w);
  }
  // wave32 reduce
  #pragma unroll
  for (int off = 16; off > 0; off >>= 1) s += __shfl_down(s, off, 32);
  __shared__ float ls[8];
  if ((threadIdx.x & 31) == 0) ls[threadIdx.x >> 5] = s;
  __syncthreads();
  if (threadIdx.x == 0) {
    float t = 0.f;
    #pragma unroll
    for (int w = 0; w < 8; ++w) t += ls[w];
    xg[j] = t * (1.0f / HW);
  }
}

// ---------------------------------------------------------------------------
// Kernel 2: quantum circuit + gate + 2-layer MLP head.  One block of 64
// threads (2 waves); LDS staging instead of private arrays so nothing spills.
// Produces add_term[b][o] = conv_b[o] + gate[b]*xgp[b][o] and writes the
// x_quantum / quantum_strength output tails.
// ---------------------------------------------------------------------------
__global__ __launch_bounds__(64) void qd_head_kernel(
    const float* __restrict__ xg, const float* __restrict__ cw,
    const float* __restrict__ qw, const float* __restrict__ w1,
    const float* __restrict__ b1, const float* __restrict__ w2,
    const float* __restrict__ b2, const float* __restrict__ conv_b,
    float* __restrict__ add_term, float* __restrict__ out) {
  __shared__ float s_xq[8];
  __shared__ float s_h[64];
  __shared__ float s_gate;
  const int t = threadIdx.x;
  const float PI = 3.14159265358979323846f;

  for (int b = 0; b < NB; ++b) {
    if (t < 4) {
      const int ia = 2 * t, ib = 2 * t + 1;
      float tha = 0.5f * (PI * xg[b * NC + ia] * cw[ia] + qw[ia]);
      float thb = 0.5f * (PI * xg[b * NC + ib] * cw[ib] + qw[ib]);
      float ca = __cosf(tha), sa = __sinf(tha);
      float cb = __cosf(thb), sb = __sinf(thb);
      // post-CNOT 2-qubit real statevector (2x2)
      float psi00 = ca * cb, psi01 = ca * sb;
      float psi10 = sa * sb, psi11 = sa * cb;
      float ha = 0.5f * qw[8 + ia], hb = 0.5f * qw[8 + ib];
      float cA = __cosf(ha), sA = __sinf(ha);
      float cB = __cosf(hb), sB = __sinf(hb);
      // psi' = Ra * psi * Rb^T
      float t00 = cA * psi00 - sA * psi10, t01 = cA * psi01 - sA * psi11;
      float t10 = sA * psi00 + cA * psi10, t11 = sA * psi01 + cA * psi11;
      float f00 = t00 * cB - t01 * sB, f01 = t00 * sB + t01 * cB;
      float f10 = t10 * cB - t11 * sB, f11 = t10 * sB + t11 * cB;
      float p00 = f00 * f00, p01 = f01 * f01, p10 = f10 * f10, p11 = f11 * f11;
      s_xq[ia] = (p00 + p01) - (p10 + p11);
      s_xq[ib] = (p00 + p10) - (p01 + p11);
    }
    __syncthreads();

    if (t == 0) {
      float ss = 0.f;
      #pragma unroll
      for (int i = 0; i < 8; ++i) ss += s_xq[i] * s_xq[i];
      float st = sqrtf(ss);
      out[OUT_MAIN + 128 + b] = st;                       // quantum_strength
      s_gate = 1.0f / (1.0f + __expf(-(st - 0.05f)));
    }
    if (t < 8) out[OUT_MAIN + b * 8 + t] = s_xq[t];       // x_quantum

    // h[t] = relu(b1[t] + sum_i xq[i]*w1[t][i])
    {
      float v = b1[t];
      #pragma unroll
      for (int i = 0; i < 8; ++i) v += s_xq[i] * w1[t * 8 + i];
      s_h[t] = v > 0.f ? v : 0.f;
    }
    __syncthreads();

    // xgp[t] = b2[t] + sum_j h[j]*w2[t][j]; fold gate + conv bias
    {
      float acc = b2[t];
      for (int j = 0; j < 64; ++j) acc += s_h[j] * w2[t * 64 + j];
      add_term[b * NC + t] = conv_b[t] + s_gate * acc;
    }
    __syncthreads();   // protect s_xq/s_h/s_gate before next batch
  }
}

// ---------------------------------------------------------------------------
// Kernel 3: 1x1 conv as fp32 WMMA GEMM, fused broadcast-add epilogue.
//   D[o][px] = sum_c conv_w[o][c] * x[b][c][px] + add_term[b][o]
// M = outputs (16 per wave), N = pixels (64 per block), K = channels (64).
// Block = 4 waves; wave w -> outputs 16w..16w+15.  16 x V_WMMA_F32_16X16X4_F32
// per accumulator tile; pixels map to the N dimension so every C/D VGPR and
// every B-matrix load covers 64B-contiguous segments per half-wave.
// A-panel pairs are loaded inside the K-loop (WGP$-resident 16KB weight
// matrix) and the loop is unrolled 4x to keep live VGPRs well under 128.
// Output stores are non-temporal so the 268MB dead write stream doesn't
// evict x from L2 between the mean pass and this pass.
// ---------------------------------------------------------------------------
__global__ __launch_bounds__(128) void qd_conv_wmma_kernel(
    const float* __restrict__ x, const float* __restrict__ conv_w,
    const float* __restrict__ add_term, float* __restrict__ out) {
  const int wave = threadIdx.x >> 5;
  const int lane = threadIdx.x & 31;
  const int half = lane >> 4;        // half-wave selects K-pair / M-high
  const int lm = lane & 15;
  const long p0 = (long)blockIdx.x * 64;     // pixel base (64 | 65536: no image straddle)
  const int b = (int)(p0 >> 16);             // image index
  const int s0 = (int)(p0 & 65535);          // spatial offset within image
  const float* xb = x + (long)b * NC * HW;
  const int obase = wave * 16;

  // A(16x4) layout: lane holds M=lm, VGPR r holds K = r + 2*half.
  const float* wr = conv_w + (obase + lm) * NC + 2 * half;
  // B(4x16) layout: lane holds N = s0+16t+lm, VGPR r holds K = r + 2*half.
  const float* xp = xb + (long)(2 * half) * HW + s0 + lm;

  v8f acc0 = {}, acc1 = {}, acc2 = {}, acc3 = {};
  #pragma unroll 4
  for (int k = 0; k < 16; ++k) {
    v2f a;
    a.x = wr[4 * k];
    a.y = wr[4 * k + 1];
    const float* xk = xp + (long)(4 * k) * HW;
    v2f b0, b1, b2, b3;
    b0.x = xk[0];  b0.y = xk[HW];
    b1.x = xk[16]; b1.y = xk[HW + 16];
    b2.x = xk[32]; b2.y = xk[HW + 32];
    b3.x = xk[48]; b3.y = xk[HW + 48];
    acc0 = __builtin_amdgcn_wmma_f32_16x16x4_f32(false, a, false, b0,
                                                 (short)0, acc0, false, false);
    acc1 = __builtin_amdgcn_wmma_f32_16x16x4_f32(false, a, false, b1,
                                                 (short)0, acc1, false, false);
    acc2 = __builtin_amdgcn_wmma_f32_16x16x4_f32(false, a, false, b2,
                                                 (short)0, acc2, false, false);
    acc3 = __builtin_amdgcn_wmma_f32_16x16x4_f32(false, a, false, b3,
                                                 (short)0, acc3, false, false);
  }

  // Epilogue: D element r at lane L is (M = r + 8*half, N = lm); add the
  // per-(b,o) broadcast term and store (64B-contiguous per half-wave).
  float am[8];
  const float* ap = add_term + b * NC + obase + 8 * half;
  #pragma unroll
  for (int r = 0; r < 8; ++r) am[r] = ap[r];

  float* ob = out + ((long)b * NC + obase + 8 * half) * HW + s0 + lm;
  #pragma unroll
  for (int r = 0; r < 8; ++r) {
    float* orow = ob + (long)r * HW;
    __builtin_nontemporal_store(acc0[r] + am[r], orow);
    __builtin_nontemporal_store(acc1[r] + am[r], orow + 16);
    __builtin_nontemporal_store(acc2[r] + am[r], orow + 32);
    __builtin_nontemporal_store(acc3[r] + am[r], orow + 48);
  }
}

// ---------------------------------------------------------------------------
extern "C" void kernel_launch(void* const* d_in, const int* in_sizes, int n_in,
                              void* d_out, int out_size, void* d_ws, size_t ws_size,
                              hipStream_t stream) {
  const float* x      = (const float*)d_in[0];
  const float* cw     = (const float*)d_in[1];
  const float* qw     = (const float*)d_in[2];
  const float* w1     = (const float*)d_in[3];
  const float* b1     = (const float*)d_in[4];
  const float* w2     = (const float*)d_in[5];
  const float* b2     = (const float*)d_in[6];
  const float* conv_w = (const float*)d_in[7];
  const float* conv_b = (const float*)d_in[8];
  float* out = (float*)d_out;

  float* xg       = (float*)d_ws;   // [16,64] spatial means
  float* add_term = xg + 1024;      // [16,64] fused epilogue term

  qd_mean_kernel<<<NB * NC, 256, 0, stream>>>(x, xg);
  qd_head_kernel<<<1, 64, 0, stream>>>(xg, cw, qw, w1, b1, w2, b2, conv_b,
                                       add_term, out);
  // 1,048,576 pixels / 64 per block = 16384 blocks of 4 waves
  qd_conv_wmma_kernel<<<16384, 128, 0, stream>>>(x, conv_w, add_term, out);
}